// NerfRaymarcher_72636486910464
// MI455X (gfx1250) — compile-verified
//
#include <hip/hip_runtime.h>
#include <cstdint>
#include <cstddef>

// ---------------------------------------------------------------------------
// NeRF raymarcher for MI455X (gfx1250).
// Bandwidth-bound streaming scan: one wave32 per ray, TDM async staging of
// per-ray tiles into LDS (double buffered), wave product-scan for the
// shifted cumprod, shfl_xor reductions, non-temporal streaming stores.
// ---------------------------------------------------------------------------

#define NPTS 128
#define EPSW 1e-10f

typedef float        v4f __attribute__((ext_vector_type(4)));
typedef float        v2f __attribute__((ext_vector_type(2)));
typedef unsigned int v4u __attribute__((ext_vector_type(4)));
typedef int          v8i __attribute__((ext_vector_type(8)));
typedef int          v4i __attribute__((ext_vector_type(4)));

#ifndef __has_builtin
#define __has_builtin(x) 0
#endif

#if __has_builtin(__builtin_amdgcn_tensor_load_to_lds) && \
    __has_builtin(__builtin_amdgcn_s_wait_tensorcnt)
#define USE_TDM 1
#else
#define USE_TDM 0
#endif

#if USE_TDM
// 1D tile copy of n_dwords 32-bit words from global 'gsrc' to LDS byte offset
// 'lds_byte_off', using the Tensor Data Mover (D# per ISA cdna5 ch.8).
__device__ __forceinline__ void tdm_load_1d(uint32_t lds_byte_off,
                                            const void* gsrc,
                                            uint32_t n_dwords) {
  uint64_t ga = (uint64_t)(uintptr_t)gsrc;
  v4u g0;
  g0.x = 1u;                                                 // count=1 (valid), user mode
  g0.y = lds_byte_off;                                       // lds_addr [63:32]
  g0.z = (uint32_t)ga;                                       // global_addr[31:0]
  g0.w = ((uint32_t)(ga >> 32) & 0x01FFFFFFu) | (2u << 30);  // addr[56:32] | type=2
  v8i g1;
  g1[0] = (int)(2u << 16);                                   // data_size=2 (4B), mask=0
  g1[1] = (int)((n_dwords & 0xFFFFu) << 16);                 // tensor_dim0[15:0]
  g1[2] = (int)(((n_dwords >> 16) & 0xFFFFu) | (1u << 16));  // tensor_dim0[31:16] | tensor_dim1=1
  g1[3] = (int)((n_dwords & 0xFFFFu) << 16);                 // tile_dim0 = n
  g1[4] = 0;                                                 // tile_dim1=0 (1D), tile_dim2=0
  g1[5] = (int)n_dwords;                                     // tensor_dim0_stride lo
  g1[6] = 0;
  g1[7] = 0;
  v4i z4 = {0, 0, 0, 0};
#if defined(__clang_major__) && __clang_major__ >= 23
  v8i z8 = {0, 0, 0, 0, 0, 0, 0, 0};
  __builtin_amdgcn_tensor_load_to_lds(g0, g1, z4, z4, z8, 0);
#else
  __builtin_amdgcn_tensor_load_to_lds(g0, g1, z4, z4, 0);
#endif
}
#endif  // USE_TDM

// Per-wave buffer layout (floats): feat[512] | dens[128] | len[128]  = 768
#define BUF_FLOATS 768
#define FEAT_OFF   0
#define DENS_OFF   512
#define LEN_OFF    640

__global__ __launch_bounds__(256) void nerf_raymarch_kernel(
    const float* __restrict__ dens,    // [N,128,1]
    const float* __restrict__ feats,   // [N,128,4]
    const float* __restrict__ lens,    // [N,128]
    const float* __restrict__ dirs,    // [N,3]
    const float* __restrict__ maxd,    // [1]
    float* __restrict__ out6,          // [N,6] {f0..f3, opacity, depth}
    float* __restrict__ outw,          // [N,128] weights
    int N) {
  // 8 waves/block * 2 buffers * 768 floats = 48 KiB (WGP has 320 KB LDS)
  __shared__ float smem[8 * 2 * BUF_FLOATS];

  const int lane   = threadIdx.x & 31;
  const int wave   = threadIdx.x >> 5;
  const int gw     = blockIdx.x * (blockDim.x >> 5) + wave;
  const int stride = gridDim.x * (blockDim.x >> 5);
  const float max_depth = maxd[0];

  auto issue = [&](int r, int parity) {
    float* buf = &smem[(wave * 2 + parity) * BUF_FLOATS];
#if USE_TDM
    uint32_t boff = (uint32_t)(uintptr_t)buf;  // low 32b of LDS flat addr = LDS offset
    tdm_load_1d(boff,                 feats + (size_t)r * 512, 512);
    tdm_load_1d(boff + DENS_OFF * 4u, dens  + (size_t)r * 128, 128);
    tdm_load_1d(boff + LEN_OFF  * 4u, lens  + (size_t)r * 128, 128);
#else
    const v4f* fs = (const v4f*)(feats + (size_t)r * 512);
    v4f* fd = (v4f*)buf;
#pragma unroll
    for (int i = 0; i < 4; ++i) fd[i * 32 + lane] = fs[i * 32 + lane];
    ((v4f*)(buf + DENS_OFF))[lane] = ((const v4f*)(dens + (size_t)r * 128))[lane];
    ((v4f*)(buf + LEN_OFF))[lane]  = ((const v4f*)(lens + (size_t)r * 128))[lane];
#endif
  };

  int r = gw;
  int parity = 0;
  if (r < N) issue(r, 0);

  while (r < N) {
    const int rn = r + stride;
    if (rn < N) {
      issue(rn, parity ^ 1);                    // prefetch next ray into other buffer
      __builtin_prefetch(dirs + (size_t)rn * 3, 0, 0);
#if USE_TDM
      __builtin_amdgcn_s_wait_tensorcnt(3);     // first 3 TDM ops (current ray) done
#endif
    } else {
#if USE_TDM
      __builtin_amdgcn_s_wait_tensorcnt(0);
#endif
    }

    const float* buf = &smem[(wave * 2 + parity) * BUF_FLOATS];

    // Lane L owns points p = 4L..4L+3 (ds_load_b128, contiguous per wave).
    v4f fv0 = ((const v4f*)(buf + FEAT_OFF))[lane * 4 + 0];
    v4f fv1 = ((const v4f*)(buf + FEAT_OFF))[lane * 4 + 1];
    v4f fv2 = ((const v4f*)(buf + FEAT_OFF))[lane * 4 + 2];
    v4f fv3 = ((const v4f*)(buf + FEAT_OFF))[lane * 4 + 3];
    v4f dv  = ((const v4f*)(buf + DENS_OFF))[lane];
    v4f lv  = ((const v4f*)(buf + LEN_OFF))[lane];

    // ray length (uniform per wave -> scalarized loads)
    const float* dp = dirs + (size_t)r * 3;
    const float dx = dp[0], dy = dp[1], dz = dp[2];
    const float rl = sqrtf(dx * dx + dy * dy + dz * dz);

    // length intervals; boundary length comes from lane+1 via shuffle
    const float nl0 = __shfl_down(lv.x, 1, 32);
    const float i0 = lv.y - lv.x;
    const float i1 = lv.z - lv.y;
    const float i2 = lv.w - lv.z;
    const float i3 = (lane == 31) ? (max_depth - lv.w) : (nl0 - lv.w);

    const float a0 = 1.0f - __expf(-dv.x * i0 * rl);
    const float a1 = 1.0f - __expf(-dv.y * i1 * rl);
    const float a2 = 1.0f - __expf(-dv.z * i2 * rl);
    const float a3 = 1.0f - __expf(-dv.w * i3 * rl);

    const float t0 = fminf(1.0f, 1.0f - a0 + EPSW);
    const float t1 = fminf(1.0f, 1.0f - a1 + EPSW);
    const float t2 = fminf(1.0f, 1.0f - a2 + EPSW);
    const float t3 = fminf(1.0f, 1.0f - a3 + EPSW);

    // in-lane exclusive prefix products, then wave32 product scan (5 steps)
    const float lp1 = t0, lp2 = t0 * t1, lp3 = lp2 * t2;
    float s = lp3 * t3;  // lane-local total
#pragma unroll
    for (int d = 1; d < 32; d <<= 1) {
      const float o = __shfl_up(s, d, 32);
      if (lane >= d) s *= o;
    }
    float excl = __shfl_up(s, 1, 32);
    if (lane == 0) excl = 1.0f;

    const float w0 = excl * a0;
    const float w1 = excl * lp1 * a1;
    const float w2 = excl * lp2 * a2;
    const float w3 = excl * lp3 * a3;

    // streaming store of weights (write-once output; keep it out of L2's way)
    v4f wv = {w0, w1, w2, w3};
    __builtin_nontemporal_store(wv, (v4f*)(outw + (size_t)r * NPTS) + lane);

    // opacity: full product of (1 - alpha)  (exact, no EPS/min, per reference)
    float q = (1.0f - a0) * (1.0f - a1) * (1.0f - a2) * (1.0f - a3);
#pragma unroll
    for (int d = 16; d >= 1; d >>= 1) q *= __shfl_xor(q, d, 32);

    // feature + depth reductions
    v4f acc = fv0 * w0 + fv1 * w1 + fv2 * w2 + fv3 * w3;
    float accd = w0 * lv.x + w1 * lv.y + w2 * lv.z + w3 * lv.w;
#pragma unroll
    for (int d = 16; d >= 1; d >>= 1) {
      acc.x += __shfl_xor(acc.x, d, 32);
      acc.y += __shfl_xor(acc.y, d, 32);
      acc.z += __shfl_xor(acc.z, d, 32);
      acc.w += __shfl_xor(acc.w, d, 32);
      accd  += __shfl_xor(accd,  d, 32);
    }

    if (lane == 0) {
      float* o = out6 + (size_t)r * 6;   // rows are 24B -> 8B-aligned stores
      v2f p01 = {acc.x, acc.y};
      v2f p23 = {acc.z, acc.w};
      v2f p45 = {1.0f - q, accd};
      __builtin_nontemporal_store(p01, (v2f*)(o + 0));
      __builtin_nontemporal_store(p23, (v2f*)(o + 2));
      __builtin_nontemporal_store(p45, (v2f*)(o + 4));
    }

    parity ^= 1;
    r = rn;
  }
}

extern "C" void kernel_launch(void* const* d_in, const int* in_sizes, int n_in,
                              void* d_out, int out_size, void* d_ws, size_t ws_size,
                              hipStream_t stream) {
  const float* dens  = (const float*)d_in[0];  // [N,128,1]
  const float* feats = (const float*)d_in[1];  // [N,128,4]
  const float* lens  = (const float*)d_in[2];  // [N,128]
  const float* dirs  = (const float*)d_in[3];  // [N,3]
  const float* maxd  = (const float*)d_in[4];  // [1]

  const int N = in_sizes[2] / NPTS;

  float* out6 = (float*)d_out;                    // [N,6] first, in return order
  float* outw = (float*)d_out + (size_t)N * 6;    // then weights [N,128]

  const dim3 block(256);            // 8 wave32 per block
  const dim3 grid(2048);            // 16384 waves -> 16 rays/wave, grid-stride

  hipLaunchKernelGGL(nerf_raymarch_kernel, grid, block, 0, stream,
                     dens, feats, lens, dirs, maxd, out6, outw, N);
}